// DifferentiableBiquadChain_20675972563313
// MI455X (gfx1250) — compile-verified
//
#include <hip/hip_runtime.h>
#include <math.h>

// CDNA5 (gfx1250) block-lifted IIR EQ cascade, rev 3.
//  - Each 2-state SVF band is linear; lift to 16-sample blocks:
//      y_blk  = T(16x16 Toeplitz of impulse resp) @ u_blk + O(16x2) @ s_start
//      s_next = A^16 @ s_start + Ctrl(2x16) @ u_blk
//    T@U via V_WMMA_F32_16X16X4_F32 (f32: IIR recurrences are precision-critical).
//  - 128-block state chain = associative affine scan, 7 LDS rounds.
//  - Zero-padded impulse response -> branchless A-frag loads; stride-17 frame
//    layout -> conflict-free LDS; in_g folded into band-0 operators; async
//    global->LDS staging (global_load_async_to_lds_b32 + s_wait_asynccnt).

#define FRAMEN   2048
#define NFRAMES  24
#define NB       16
#define NBLK     128
#define BLKL     16
#define NTHREADS 128
#define XPAD(n)  ((n) + ((n) >> 4))       /* stride-17 block layout */
#define XBUF     (FRAMEN + FRAMEN / 16)   /* 2176 floats */
#define OPS      112                      /* floats per band */

#if defined(__has_builtin)
#if __has_builtin(__builtin_amdgcn_global_load_async_to_lds_b32) && \
    __has_builtin(__builtin_amdgcn_s_wait_asynccnt)
#define USE_ASYNC_LDS 1
#else
#define USE_ASYNC_LDS 0
#endif
#else
#define USE_ASYNC_LDS 0
#endif

typedef __attribute__((ext_vector_type(2))) float v2f;
typedef __attribute__((ext_vector_type(8))) float v8f;
typedef __attribute__((address_space(1))) int gint_t;
typedef __attribute__((address_space(3))) int lint_t;

__global__ __launch_bounds__(NTHREADS)
void eq_chain_kernel(const float* __restrict__ audio,
                     const float* __restrict__ params,
                     float* __restrict__ out)
{
  // per band: [0..31] hpad (first 16 zero), [32..63] O, [64..95] Ctrl, [96..99] A16
  __shared__ float sOps[NB * OPS];
  __shared__ float sX0[XBUF];
  __shared__ float sX1[XBUF];
  __shared__ float sScanM[NBLK * 4];
  __shared__ float sScanT[NBLK * 2];
  __shared__ float sBlk[NBLK * 2];
  __shared__ float sState[NB * 2];
  __shared__ float sGain[1];

  const int b    = blockIdx.x;
  const int tid  = threadIdx.x;
  const int lane = tid & 31;
  const int rowA = lane & 15;   // A-frag row / B-frag column (N)
  const int hi   = lane >> 4;   // half-wave K split per ISA VGPR layout
  const int wv   = tid >> 5;

  const float* __restrict__ Pb = params + (size_t)b * 50 * NFRAMES;
  const float* __restrict__ Ab = audio  + (size_t)b * NFRAMES * FRAMEN;
  float*       __restrict__ Ob = out    + (size_t)b * NFRAMES * FRAMEN;

  if (tid < NB * 2) sState[tid] = 0.f;

  for (int f = 0; f < NFRAMES; ++f) {
    __syncthreads();

    // ---- per-frame operator build: thread `band` builds its band ----
    if (tid < NB) {
      const int band = tid;
      const float fn = Pb[(3 * band + 0) * NFRAMES + f];
      const float gn = Pb[(3 * band + 1) * NFRAMES + f];
      const float qn = Pb[(3 * band + 2) * NFRAMES + f];

      float Q = expf(-0.69314718f + qn * 3.46573590f);     // [0.5,16]
      Q = fminf(fmaxf(Q, 0.1f), 100.f);
      const float gain = -24.f + gn * 48.f;
      const float Aa = expf(gain * 0.05756462732f);        // 10^(gain/40)
      const float sA = sqrtf(Aa);

      float a1, a2, a3, m0, m1, m2;
      if (band == 0) {            // highpass, 20..500 Hz
        float fc = expf(2.99573227f + fn * (6.21460810f - 2.99573227f));
        float g  = fminf(fmaxf(tanf(3.14159265f * fc / 96000.f), 1e-6f), 100.f);
        float k  = 1.f / Q;
        a1 = 1.f / (1.f + g * (g + k)); a2 = g * a1; a3 = g * a2;
        m0 = 1.f; m1 = -k; m2 = -1.f;
      } else if (band == 15) {    // lowpass, 5000..20000 Hz
        float fc = expf(8.51719319f + fn * (9.90348755f - 8.51719319f));
        float g  = fminf(fmaxf(tanf(3.14159265f * fc / 96000.f), 1e-6f), 100.f);
        float k  = 1.f / Q;
        a1 = 1.f / (1.f + g * (g + k)); a2 = g * a1; a3 = g * a2;
        m0 = 0.f; m1 = 0.f; m2 = 1.f;
      } else if (band == 1) {     // lowshelf, 50..16000 Hz
        float fc = expf(3.91202301f + fn * (9.68034398f - 3.91202301f));
        float g  = fminf(fmaxf(tanf(3.14159265f * fc / 96000.f), 1e-6f), 100.f);
        float k  = 1.f / Q;
        float gs = (gain >= 0.f) ? g / sA : g * sA;
        a1 = 1.f / (1.f + gs * (gs + k)); a2 = gs * a1; a3 = gs * a2;
        m0 = 1.f; m1 = k * (Aa - 1.f); m2 = Aa * Aa - 1.f;
      } else if (band == 14) {    // highshelf, 50..16000 Hz
        float fc = expf(3.91202301f + fn * (9.68034398f - 3.91202301f));
        float g  = fminf(fmaxf(tanf(3.14159265f * fc / 96000.f), 1e-6f), 100.f);
        float k  = 1.f / Q;
        float gs = (gain >= 0.f) ? g * sA : g / sA;
        a1 = 1.f / (1.f + gs * (gs + k)); a2 = gs * a1; a3 = gs * a2;
        m0 = Aa * Aa; m1 = k * (1.f - Aa) * Aa; m2 = 1.f - Aa * Aa;
      } else {                    // peak, 100..15000 Hz
        float fc = expf(4.60517019f + fn * (9.61580548f - 4.60517019f));
        float g  = fminf(fmaxf(tanf(3.14159265f * fc / 96000.f), 1e-6f), 100.f);
        float k  = (gain >= 0.f) ? 1.f / (Q * Aa) : Aa / Q;
        a1 = 1.f / (1.f + g * (g + k)); a2 = g * a1; a3 = g * a2;
        m0 = 1.f; m1 = k * (Aa * Aa - 1.f); m2 = 0.f;
      }

      // per-sample affine map:  s' = A s + B x ; y = C s + D x
      const float q23 = a2 * a2 + a3;
      const float A00 = 2.f * a1 - 1.f, A01 = -2.f * a2;
      const float A10 = 2.f * a1 * a2,  A11 = 1.f - 2.f * q23;
      float B0v = 2.f * a2,             B1v = 2.f * q23;
      const float C0v = m1 * a1 + m2 * a1 * a2;
      const float C1v = -m1 * a2 + m2 * (1.f - q23);
      float Dv  = m0 + m1 * a2 + m2 * q23;

      // fold broadband input gain into band-0 (linearity: scale B and D)
      if (band == 0) {
        float pin = Pb[48 * NFRAMES + f];
        float ing = expf((-60.f + pin * 60.f) * 0.11512925465f);  // 10^(g/20)
        B0v *= ing; B1v *= ing; Dv *= ing;
      }

      float* op = sOps + band * OPS;
#pragma unroll
      for (int z = 0; z < 16; z++) op[z] = 0.f;   // zero pad -> branchless T
      float g0[16], g1[16];
      op[16] = Dv;                      // h[0]
      g0[0] = B0v; g1[0] = B1v;         // g_j = A^j B
#pragma unroll
      for (int k2 = 1; k2 < 16; k2++) {
        op[16 + k2] = C0v * g0[k2 - 1] + C1v * g1[k2 - 1]; // h[k] = C A^{k-1} B
        g0[k2] = A00 * g0[k2 - 1] + A01 * g1[k2 - 1];
        g1[k2] = A10 * g0[k2 - 1] + A11 * g1[k2 - 1];
      }
      float r0 = C0v, r1 = C1v;         // O[r] = C A^r
      op[32] = r0; op[33] = r1;
#pragma unroll
      for (int r = 1; r < 16; r++) {
        float n0 = r0 * A00 + r1 * A10;
        float n1 = r0 * A01 + r1 * A11;
        op[32 + 2 * r] = n0; op[33 + 2 * r] = n1;
        r0 = n0; r1 = n1;
      }
#pragma unroll
      for (int kc = 0; kc < 16; kc++) { // Ctrl[:,k] = A^{15-k} B
        op[64 + 2 * kc]     = g0[15 - kc];
        op[64 + 2 * kc + 1] = g1[15 - kc];
      }
      float M00 = A00, M01 = A01, M10 = A10, M11 = A11;   // A^16
#pragma unroll
      for (int sq = 0; sq < 4; sq++) {
        float t00 = M00 * M00 + M01 * M10, t01 = M00 * M01 + M01 * M11;
        float t10 = M10 * M00 + M11 * M10, t11 = M10 * M01 + M11 * M11;
        M00 = t00; M01 = t01; M10 = t10; M11 = t11;
      }
      op[96] = M00; op[97] = M01; op[98] = M10; op[99] = M11;
    } else if (tid == NB) {
      float pot = Pb[49 * NFRAMES + f];
      sGain[0] = expf((-60.f + pot * 60.f) * 0.11512925465f);
    }

    // ---- stage raw frame into LDS (padded layout) ----
#if USE_ASYNC_LDS
#pragma unroll
    for (int i = 0; i < FRAMEN / NTHREADS; i++) {
      int n = tid + NTHREADS * i;
      __builtin_amdgcn_global_load_async_to_lds_b32(
          (gint_t*)(void*)(Ab + f * FRAMEN + n),
          (lint_t*)(sX0 + XPAD(n)),
          0, 0);
    }
    __builtin_amdgcn_s_wait_asynccnt(0);
#else
#pragma unroll
    for (int i = 0; i < FRAMEN / NTHREADS; i++) {
      int n = tid + NTHREADS * i;
      sX0[XPAD(n)] = Ab[f * FRAMEN + n];
    }
#endif
    if (f + 1 < NFRAMES)
      __builtin_prefetch(&Ab[(f + 1) * FRAMEN + tid * 16], 0, 0); // global_prefetch_b8
    __syncthreads();

    float* cur = sX0;
    float* nxt = sX1;

    for (int band = 0; band < NB; ++band) {
      const float* op  = sOps + band * OPS;
      const float  s0x = sState[2 * band], s0y = sState[2 * band + 1];

      // ---- Phase A: P = T @ U via 16x16x4 f32 WMMA (2 col-tiles / wave) ----
      v8f acc0 = {0, 0, 0, 0, 0, 0, 0, 0};
      v8f acc1 = {0, 0, 0, 0, 0, 0, 0, 0};
      const int kb0 = (2 * wv) * BLKL + rowA;       // block (= U column) index
      const int kb1 = (2 * wv + 1) * BLKL + rowA;
#pragma unroll
      for (int j = 0; j < 4; j++) {
        // A-frag: T[m][k] = hpad[16+m-k]; layout: v0 = K 0/2, v1 = K 1/3
        const int k0 = 4 * j + 2 * hi;
        v2f a;
        a.x = op[16 + rowA - k0];
        a.y = op[16 + rowA - k0 - 1];
        // B-frag: U[K][c] = x[17c + K] (stride-17, bank-conflict-free)
        const int kk = 4 * j + hi;
        v2f bb0, bb1;
        bb0.x = cur[17 * kb0 + kk];  bb0.y = cur[17 * kb0 + kk + 2];
        bb1.x = cur[17 * kb1 + kk];  bb1.y = cur[17 * kb1 + kk + 2];
        acc0 = __builtin_amdgcn_wmma_f32_16x16x4_f32(false, a, false, bb0,
                                                     (short)0, acc0, false, false);
        acc1 = __builtin_amdgcn_wmma_f32_16x16x4_f32(false, a, false, bb1,
                                                     (short)0, acc1, false, false);
      }

      // ---- Phase B: affine scan over 128 blocks: s' = A16 s + Ctrl u ----
      float eM00 = op[96], eM01 = op[97], eM10 = op[98], eM11 = op[99];
      float eT0 = 0.f, eT1 = 0.f;
      {
        const float* ctrl = op + 64;
#pragma unroll
        for (int jj = 0; jj < 16; jj++) {
          float u = cur[17 * tid + jj];
          eT0 = fmaf(ctrl[2 * jj], u, eT0);
          eT1 = fmaf(ctrl[2 * jj + 1], u, eT1);
        }
      }
      sScanM[4 * tid] = eM00; sScanM[4 * tid + 1] = eM01;
      sScanM[4 * tid + 2] = eM10; sScanM[4 * tid + 3] = eM11;
      sScanT[2 * tid] = eT0; sScanT[2 * tid + 1] = eT1;
      __syncthreads();
      for (int d = 1; d < NBLK; d <<= 1) {
        float oM00, oM01, oM10, oM11, oT0, oT1;
        const bool hasp = (tid >= d);
        if (hasp) {
          int p = tid - d;
          oM00 = sScanM[4 * p];     oM01 = sScanM[4 * p + 1];
          oM10 = sScanM[4 * p + 2]; oM11 = sScanM[4 * p + 3];
          oT0 = sScanT[2 * p];      oT1 = sScanT[2 * p + 1];
        }
        __syncthreads();
        if (hasp) {   // mine (newer) on the left: e = e_mine ∘ e_other
          float n00 = eM00 * oM00 + eM01 * oM10;
          float n01 = eM00 * oM01 + eM01 * oM11;
          float n10 = eM10 * oM00 + eM11 * oM10;
          float n11 = eM10 * oM01 + eM11 * oM11;
          float nt0 = eM00 * oT0 + eM01 * oT1 + eT0;
          float nt1 = eM10 * oT0 + eM11 * oT1 + eT1;
          eM00 = n00; eM01 = n01; eM10 = n10; eM11 = n11; eT0 = nt0; eT1 = nt1;
          sScanM[4 * tid] = eM00; sScanM[4 * tid + 1] = eM01;
          sScanM[4 * tid + 2] = eM10; sScanM[4 * tid + 3] = eM11;
          sScanT[2 * tid] = eT0; sScanT[2 * tid + 1] = eT1;
        }
        __syncthreads();
      }
      // state at start of my block = exclusive prefix applied to frame state
      float sk0, sk1;
      if (tid == 0) { sk0 = s0x; sk1 = s0y; }
      else {
        int p = tid - 1;
        float pM00 = sScanM[4 * p], pM01 = sScanM[4 * p + 1];
        float pM10 = sScanM[4 * p + 2], pM11 = sScanM[4 * p + 3];
        float pT0 = sScanT[2 * p], pT1 = sScanT[2 * p + 1];
        sk0 = pM00 * s0x + pM01 * s0y + pT0;
        sk1 = pM10 * s0x + pM11 * s0y + pT1;
      }
      sBlk[2 * tid] = sk0; sBlk[2 * tid + 1] = sk1;
      if (tid == NBLK - 1) {      // carry band state to next frame
        sState[2 * band]     = eM00 * s0x + eM01 * s0y + eT0;
        sState[2 * band + 1] = eM10 * s0x + eM11 * s0y + eT1;
      }
      __syncthreads();

      // ---- Phase C: y = P + O @ s_blk ; write next-band buffer ----
      {
        const float* Orow = op + 32;
        float sb0 = sBlk[2 * kb0], sb1 = sBlk[2 * kb0 + 1];
#pragma unroll
        for (int vr = 0; vr < 8; vr++) {
          int M = vr + 8 * hi;    // C/D layout: M = vgpr (+8 upper half-wave)
          nxt[17 * kb0 + M] = acc0[vr] + Orow[2 * M] * sb0 + Orow[2 * M + 1] * sb1;
        }
        sb0 = sBlk[2 * kb1]; sb1 = sBlk[2 * kb1 + 1];
#pragma unroll
        for (int vr = 0; vr < 8; vr++) {
          int M = vr + 8 * hi;
          nxt[17 * kb1 + M] = acc1[vr] + Orow[2 * M] * sb0 + Orow[2 * M + 1] * sb1;
        }
      }
      __syncthreads();
      float* t = cur; cur = nxt; nxt = t;
    }

    // ---- store frame output with out_g (16 even swaps -> cur == sX0) ----
    const float outg = sGain[0];
#pragma unroll
    for (int i = 0; i < FRAMEN / NTHREADS; i++) {
      int n = tid + NTHREADS * i;
      Ob[f * FRAMEN + n] = cur[XPAD(n)] * outg;
    }
  }
}

extern "C" void kernel_launch(void* const* d_in, const int* in_sizes, int n_in,
                              void* d_out, int out_size, void* d_ws, size_t ws_size,
                              hipStream_t stream) {
  (void)in_sizes; (void)n_in; (void)out_size; (void)d_ws; (void)ws_size;
  const float* audio  = (const float*)d_in[0];
  const float* params = (const float*)d_in[1];
  float* out = (float*)d_out;
  eq_chain_kernel<<<32, NTHREADS, 0, stream>>>(audio, params, out);
}